// TraitsPredictor_8555574853745
// MI455X (gfx1250) — compile-verified
//
#include <hip/hip_runtime.h>
#include <hip/hip_bf16.h>
#include <math.h>

#define HID 128
#define OUTD 10

typedef float v2f __attribute__((ext_vector_type(2)));
typedef float v8f __attribute__((ext_vector_type(8)));

// ---------- order-preserving float <-> uint encoding for atomicMax ----------
__device__ __forceinline__ unsigned fenc(float f) {
  unsigned u = __float_as_uint(f);
  return (u & 0x80000000u) ? ~u : (u | 0x80000000u);
}
__device__ __forceinline__ float fdec(unsigned u) {
  return (u & 0x80000000u) ? __uint_as_float(u & 0x7fffffffu) : __uint_as_float(~u);
}

// ---------- generic 32-bit fill ----------
__global__ void k_fill_u32(unsigned* __restrict__ p, unsigned v, int n) {
  int i = blockIdx.x * blockDim.x + threadIdx.x;
  if (i < n) p[i] = v;
}

// ---------- c = dot(wedge[0:128], aedge[0:128]) ----------
__global__ void k_dot128(const float* __restrict__ a, const float* __restrict__ b,
                         float* __restrict__ out) {
  __shared__ float sd[128];
  int t = threadIdx.x;
  sd[t] = a[t] * b[t];
  __syncthreads();
  for (int o = 64; o; o >>= 1) {
    if (t < o) sd[t] += sd[t + o];
    __syncthreads();
  }
  if (t == 0) out[0] = sd[0];
}

// ---------- deterministic mean(edge_attr) : two pass ----------
__global__ void k_partial_sum(const float* __restrict__ x, int n, float* __restrict__ red) {
  float a = 0.f;
  for (int i = blockIdx.x * blockDim.x + threadIdx.x; i < n; i += gridDim.x * blockDim.x)
    a += x[i];
  __shared__ float sd[256];
  sd[threadIdx.x] = a;
  __syncthreads();
  for (int o = 128; o; o >>= 1) {
    if (threadIdx.x < o) sd[threadIdx.x] += sd[threadIdx.x + o];
    __syncthreads();
  }
  if (threadIdx.x == 0) red[blockIdx.x] = sd[0];
}
__global__ void k_final_mean(const float* __restrict__ red, int n, float* __restrict__ out) {
  __shared__ float sd[256];
  sd[threadIdx.x] = red[threadIdx.x];
  __syncthreads();
  for (int o = 128; o; o >>= 1) {
    if (threadIdx.x < o) sd[threadIdx.x] += sd[threadIdx.x + o];
    __syncthreads();
  }
  if (threadIdx.x == 0) out[0] = sd[0] / (float)n;
}

// ---------- zero-pad W[128 x fin] -> Wp[128 x finp] ----------
__global__ void k_pad_w(const float* __restrict__ W, float* __restrict__ Wp, int fin,
                        int finp) {
  int idx = blockIdx.x * blockDim.x + threadIdx.x;
  if (idx >= HID * finp) return;
  int r = idx / finp, j = idx - r * finp;
  Wp[idx] = (j < fin) ? W[(size_t)r * fin + j] : 0.f;
}

// ---------- H[n x 128] = X[n x finp] @ W^T, W is [128 x finp] row-major ----------
// finp multiple of 8; X/W rows zero-padded. fp32 WMMA 16x16x4; one block = 16-row
// node tile, 8 waves cover 128 output columns. Branch-free software-pipelined K loop.
__global__ void k_gemm_wmma_f32(const float* __restrict__ X, const float* __restrict__ W,
                                float* __restrict__ H, int n, int finp) {
  const int wave = threadIdx.x >> 5;
  const int lane = threadIdx.x & 31;
  const int half = lane >> 4;   // 0 | 1
  const int l16  = lane & 15;
  const int m0   = blockIdx.x * 16;
  int m = m0 + l16;
  m = (m < n) ? m : (n - 1);          // clamp: loads always valid, stores guarded
  const int col  = wave * 16 + l16;   // B column this lane feeds
  const int koff = half * 2;          // K sub-offset per ISA fragment layout
  const float* ap = X + (size_t)m * finp + koff;
  const float* bp = W + (size_t)col * finp + koff;

  v8f acc0 = {};
  v8f acc1 = {};
  v2f a0 = *(const v2f*)(ap);
  v2f a1 = *(const v2f*)(ap + 4);
  v2f b0 = *(const v2f*)(bp);
  v2f b1 = *(const v2f*)(bp + 4);
  int kk = 8;
  for (; kk < finp; kk += 8) {
    v2f na0 = *(const v2f*)(ap + kk);
    v2f na1 = *(const v2f*)(ap + kk + 4);
    v2f nb0 = *(const v2f*)(bp + kk);
    v2f nb1 = *(const v2f*)(bp + kk + 4);
    acc0 = __builtin_amdgcn_wmma_f32_16x16x4_f32(false, a0, false, b0, (short)0, acc0,
                                                 false, false);
    acc1 = __builtin_amdgcn_wmma_f32_16x16x4_f32(false, a1, false, b1, (short)0, acc1,
                                                 false, false);
    a0 = na0; a1 = na1; b0 = nb0; b1 = nb1;
  }
  acc0 = __builtin_amdgcn_wmma_f32_16x16x4_f32(false, a0, false, b0, (short)0, acc0,
                                               false, false);
  acc1 = __builtin_amdgcn_wmma_f32_16x16x4_f32(false, a1, false, b1, (short)0, acc1,
                                               false, false);
  v8f acc = acc0 + acc1;
  const int rbase = m0 + half * 8;
#pragma unroll
  for (int r = 0; r < 8; ++r) {
    const int row = rbase + r;
    if (row < n) H[(size_t)row * HID + col] = acc[r];
  }
}

// ---------- per-node attention scalars: as = h.asrc, ad = h.adst ----------
__global__ void k_node_scalars(const float* __restrict__ H, const float* __restrict__ asrc,
                               const float* __restrict__ adst, float* __restrict__ AS,
                               float* __restrict__ AD, int n) {
  const int gw = (blockIdx.x * blockDim.x + threadIdx.x) >> 5;
  const int lane = threadIdx.x & 31;
  if (gw >= n) return;
  float4 h = ((const float4*)(H + (size_t)gw * HID))[lane];
  float4 s = ((const float4*)asrc)[lane];
  float pa = h.x * s.x + h.y * s.y + h.z * s.z + h.w * s.w;
  float pd = 0.f;
  if (adst) {
    float4 d4 = ((const float4*)adst)[lane];
    pd = h.x * d4.x + h.y * d4.y + h.z * d4.z + h.w * d4.w;
  }
  for (int o = 16; o; o >>= 1) {
    pa += __shfl_xor(pa, o, 32);
    pd += __shfl_xor(pd, o, 32);
  }
  if (lane == 0) {
    AS[gw] = pa;
    if (adst) AD[gw] = pd;
  }
}

// ---------- edge logits + running segment max ----------
__global__ void k_edge_logit(const int* __restrict__ src, const int* __restrict__ dst,
                             const float* __restrict__ ea, const float* __restrict__ AS,
                             const float* __restrict__ AD, const float* __restrict__ SC,
                             float* __restrict__ elog, unsigned* __restrict__ mx, int E) {
  int e = blockIdx.x * blockDim.x + threadIdx.x;
  if (e >= E) return;
  int s = src[e], d = dst[e];
  float l = AS[s] + (AD ? AD[d] : 0.f) + SC[0] * ea[e];
  l = (l > 0.f) ? l : 0.2f * l;  // leaky_relu(0.2)
  elog[e] = l;
  atomicMax(&mx[d], fenc(l));
}
__global__ void k_self_logit(const float* __restrict__ AS, const float* __restrict__ AD,
                             const float* __restrict__ SC, float* __restrict__ slog,
                             unsigned* __restrict__ mx, int n) {
  int i = blockIdx.x * blockDim.x + threadIdx.x;
  if (i >= n) return;
  float l = AS[i] + AD[i] + SC[0] * SC[1];  // SC[1] = mean edge attr (self-loop fill)
  l = (l > 0.f) ? l : 0.2f * l;
  slog[i] = l;
  unsigned en = fenc(l);  // edge pass already complete -> plain max
  if (en > mx[i]) mx[i] = en;
}

// ---------- exp + segment sum ----------
__global__ void k_edge_exp(const int* __restrict__ dst, float* __restrict__ elog,
                           const unsigned* __restrict__ mx, float* __restrict__ den, int E) {
  int e = blockIdx.x * blockDim.x + threadIdx.x;
  if (e >= E) return;
  int d = dst[e];
  float ex = expf(elog[e] - fdec(mx[d]));
  elog[e] = ex;
  atomicAdd(&den[d], ex);
}
__global__ void k_self_exp(float* __restrict__ slog, const unsigned* __restrict__ mx,
                           float* __restrict__ den, int n) {
  int i = blockIdx.x * blockDim.x + threadIdx.x;
  if (i >= n) return;
  float ex = expf(slog[i] - fdec(mx[i]));
  slog[i] = ex;
  den[i] += ex;  // edge pass complete -> plain add
}

// ---------- weighted scatter aggregation: out[d] += a * h[s] (wave per edge) ----------
__global__ void k_agg_edges(const int* __restrict__ src, const int* __restrict__ dst,
                            const float* __restrict__ ex, const float* __restrict__ den,
                            const float* __restrict__ H, float* __restrict__ out, int E) {
  const int e = (blockIdx.x * blockDim.x + threadIdx.x) >> 5;
  const int lane = threadIdx.x & 31;
  if (e >= E) return;
  const int s = src[e], d = dst[e];
  const float a = ex[e] / (den[d] + 1e-16f);
  float4 v = ((const float4*)(H + (size_t)s * HID))[lane];
  float* o = out + (size_t)d * HID + lane * 4;
  atomicAdd(o + 0, a * v.x);
  atomicAdd(o + 1, a * v.y);
  atomicAdd(o + 2, a * v.z);
  atomicAdd(o + 3, a * v.w);
}
__global__ void k_agg_self(const float* __restrict__ slog, const float* __restrict__ den,
                           const float* __restrict__ H, float* __restrict__ out, int n) {
  const int i = (blockIdx.x * blockDim.x + threadIdx.x) >> 5;
  const int lane = threadIdx.x & 31;
  if (i >= n) return;
  const float a = slog[i] / (den[i] + 1e-16f);
  float4 v = ((const float4*)(H + (size_t)i * HID))[lane];
  float4* o = (float4*)(out + (size_t)i * HID) + lane;
  float4 t = *o;
  t.x += a * v.x; t.y += a * v.y; t.z += a * v.z; t.w += a * v.w;
  *o = t;
}

// ---------- bias (+ optional relu) ----------
__global__ void k_finish(float* __restrict__ out, const float* __restrict__ b, int relu, int n) {
  int idx = blockIdx.x * blockDim.x + threadIdx.x;
  if (idx >= n * HID) return;
  float v = out[idx] + b[idx & (HID - 1)];
  out[idx] = relu ? fmaxf(v, 0.f) : v;
}

// ---------- concats ----------
// spatial input: 18 real columns zero-padded to 24 for branch-free WMMA K loop
__global__ void k_concat_sp(const float* __restrict__ xs, const float* __restrict__ gd,
                            float* __restrict__ o, int n) {
  int idx = blockIdx.x * blockDim.x + threadIdx.x;
  if (idx >= n * 24) return;
  int i = idx / 24, j = idx - i * 24;
  float v = 0.f;
  if (j < 16)      v = xs[(size_t)i * 16 + j];
  else if (j < 18) v = gd[(size_t)i * 2 + (j - 16)];
  o[idx] = v;
}
__global__ void k_concat_pc(const float* __restrict__ s2p, const float* __restrict__ xsp,
                            const float* __restrict__ phy, float* __restrict__ o, int n) {
  int idx = blockIdx.x * blockDim.x + threadIdx.x;
  if (idx >= n * 224) return;
  int i = idx / 224, j = idx - i * 224;
  float v;
  if (j < 128)      v = s2p[(size_t)i * 128 + j];
  else if (j < 160) v = xsp[(size_t)i * 32 + (j - 128)];
  else              v = phy[(size_t)i * 64 + (j - 160)];
  o[idx] = v;
}

// ---------- final FC: out[i,o] = emb[i]·fc_w[o] + fc_b[o] ----------
__global__ void k_fc(const float* __restrict__ emb, const float* __restrict__ w,
                     const float* __restrict__ b, float* __restrict__ out, int n) {
  int idx = blockIdx.x * blockDim.x + threadIdx.x;
  if (idx >= n * OUTD) return;
  int i = idx / OUTD, o = idx - i * OUTD;
  const float4* e = (const float4*)(emb + (size_t)i * HID);
  const float4* ww = (const float4*)(w + (size_t)o * HID);
  float acc = b[o];
#pragma unroll
  for (int j = 0; j < HID / 4; ++j) {
    float4 ev = e[j], wv = ww[j];
    acc += ev.x * wv.x + ev.y * wv.y + ev.z * wv.z + ev.w * wv.w;
  }
  out[idx] = acc;
}

// ======================= host-side orchestration =======================
struct Gat {
  const float *w, *b, *asrc, *adst, *aedge, *wedge;
  int fin;
};

static inline int cdiv(int a, int b) { return (a + b - 1) / b; }

static void run_gat(hipStream_t s, const float* X, int n, const Gat& p, const int* esrc,
                    const int* edst, const float* ea, int E, bool selfloop, int ndst,
                    float* OUT, int do_relu, float* Hb, float* AS, float* AD, unsigned* MX,
                    float* DEN, float* ELOG, float* SLOG, float* RED, float* SC,
                    float* WPAD) {
  k_dot128<<<1, 128, 0, s>>>(p.wedge, p.aedge, SC);  // SC[0] = wedge·aedge
  if (selfloop) {
    k_partial_sum<<<256, 256, 0, s>>>(ea, E, RED);
    k_final_mean<<<1, 256, 0, s>>>(RED, E, SC + 1);  // SC[1] = mean(ea)
  }
  const int finp = (p.fin + 7) & ~7;
  const float* Wuse = p.w;
  if (finp != p.fin) {  // zero-pad weights once so the WMMA K loop is branch-free
    k_pad_w<<<cdiv(HID * finp, 256), 256, 0, s>>>(p.w, WPAD, p.fin, finp);
    Wuse = WPAD;
  }
  k_gemm_wmma_f32<<<cdiv(n, 16), 256, 0, s>>>(X, Wuse, Hb, n, finp);
  k_node_scalars<<<cdiv(n, 8), 256, 0, s>>>(Hb, p.asrc, p.adst, AS, AD, n);
  k_fill_u32<<<cdiv(ndst, 256), 256, 0, s>>>(MX, 0u, ndst);
  k_fill_u32<<<cdiv(ndst, 256), 256, 0, s>>>((unsigned*)DEN, 0u, ndst);
  k_fill_u32<<<cdiv(ndst * HID, 256), 256, 0, s>>>((unsigned*)OUT, 0u, ndst * HID);
  k_edge_logit<<<cdiv(E, 256), 256, 0, s>>>(esrc, edst, ea, AS, p.adst ? AD : nullptr, SC,
                                            ELOG, MX, E);
  if (selfloop) k_self_logit<<<cdiv(n, 256), 256, 0, s>>>(AS, AD, SC, SLOG, MX, n);
  k_edge_exp<<<cdiv(E, 256), 256, 0, s>>>(edst, ELOG, MX, DEN, E);
  if (selfloop) k_self_exp<<<cdiv(n, 256), 256, 0, s>>>(SLOG, MX, DEN, n);
  k_agg_edges<<<cdiv(E, 8), 256, 0, s>>>(esrc, edst, ELOG, DEN, Hb, OUT, E);
  if (selfloop) k_agg_self<<<cdiv(n, 8), 256, 0, s>>>(SLOG, DEN, Hb, OUT, n);
  k_finish<<<cdiv(ndst * HID, 256), 256, 0, s>>>(OUT, p.b, do_relu, ndst);
}

extern "C" void kernel_launch(void* const* d_in, const int* in_sizes, int n_in, void* d_out,
                              int out_size, void* d_ws, size_t ws_size, hipStream_t stream) {
  auto F = [&](int i) { return (const float*)d_in[i]; };
  auto I = [&](int i) { return (const int*)d_in[i]; };

  // positional inputs (setup_inputs dict order)
  const float* x_spatial = F(0);
  const float* global_data = F(1);
  const int* ei_sp = I(2);  const int ES = in_sizes[2] / 2;
  const float* ea_sp = F(3);
  const int* ei_bip = I(4); const int EB = in_sizes[4] / 2;
  const float* ea_bip = F(5);
  const float* x_species = F(6);
  const float* x_phylo = F(7);
  const int* ei_pc = I(8);  const int EP = in_sizes[8] / 2;
  const float* ea_pc = F(9);
  const int NS = in_sizes[0] / 16;
  const int NP = in_sizes[6] / 32;

  // params pytree, sorted-key flatten:
  // bip{aedge,asrc,b,w,wedge}=10..14, fc_b=15, fc_w=16,
  // pc0{adst,aedge,asrc,b,w,wedge}=17..22, pc1=23..28, pc2=29..34,
  // sp0=35..40, sp1=41..46, sp2=47..52
  Gat bip = {F(13), F(12), F(11), nullptr, F(10), F(14), in_sizes[13] / HID};
  const float* fc_b = F(15);
  const float* fc_w = F(16);
  auto gat6 = [&](int base) {
    Gat g = {F(base + 4), F(base + 3), F(base + 2), F(base + 0), F(base + 1), F(base + 5),
             in_sizes[base + 4] / HID};
    return g;
  };
  Gat pc0 = gat6(17), pc1 = gat6(23), pc2 = gat6(29);
  Gat sp0 = gat6(35), sp1 = gat6(41), sp2 = gat6(47);

  // ---- workspace arena ----
  const size_t FSZ = (size_t)NS * HID;  // also covers NS*24 and NP*224
  float* F1   = (float*)d_ws;
  float* F2   = F1 + FSZ;
  float* Hb   = F2 + FSZ;
  float* ELOG = Hb + FSZ;
  float* AS   = ELOG + (size_t)ES;  // ES is max edge count
  float* AD   = AS + NS;
  unsigned* MX = (unsigned*)(AD + NS);
  float* DEN  = (float*)(MX + NS);
  float* SLOG = DEN + NS;
  float* S2P  = SLOG + NS;
  float* RED  = S2P + (size_t)NP * HID;
  float* SC   = RED + 256;
  float* WPAD = SC + 16;  // up to 128 x 256 padded weights
  (void)ws_size; (void)n_in; (void)out_size;

  // ---- spatial tower ----
  k_concat_sp<<<cdiv(NS * 24, 256), 256, 0, stream>>>(x_spatial, global_data, F1, NS);
  run_gat(stream, F1, NS, sp0, ei_sp, ei_sp + ES, ea_sp, ES, true, NS, F2, 1,
          Hb, AS, AD, MX, DEN, ELOG, SLOG, RED, SC, WPAD);
  run_gat(stream, F2, NS, sp1, ei_sp, ei_sp + ES, ea_sp, ES, true, NS, F1, 1,
          Hb, AS, AD, MX, DEN, ELOG, SLOG, RED, SC, WPAD);
  run_gat(stream, F1, NS, sp2, ei_sp, ei_sp + ES, ea_sp, ES, true, NS, F2, 1,
          Hb, AS, AD, MX, DEN, ELOG, SLOG, RED, SC, WPAD);  // F2 = relu(space_emb)

  // ---- bipartite spatial -> species ----
  run_gat(stream, F2, NS, bip, ei_bip, ei_bip + EB, ea_bip, EB, false, NP, S2P, 0,
          Hb, AS, AD, MX, DEN, ELOG, SLOG, RED, SC, WPAD);

  // ---- species tower ----
  k_concat_pc<<<cdiv(NP * 224, 256), 256, 0, stream>>>(S2P, x_species, x_phylo, F1, NP);
  run_gat(stream, F1, NP, pc0, ei_pc, ei_pc + EP, ea_pc, EP, true, NP, F2, 1,
          Hb, AS, AD, MX, DEN, ELOG, SLOG, RED, SC, WPAD);
  run_gat(stream, F2, NP, pc1, ei_pc, ei_pc + EP, ea_pc, EP, true, NP, F1, 1,
          Hb, AS, AD, MX, DEN, ELOG, SLOG, RED, SC, WPAD);
  run_gat(stream, F1, NP, pc2, ei_pc, ei_pc + EP, ea_pc, EP, true, NP, F2, 1,
          Hb, AS, AD, MX, DEN, ELOG, SLOG, RED, SC, WPAD);  // F2 = relu(species_emb)

  // ---- FC head ----
  k_fc<<<cdiv(NP * OUTD, 256), 256, 0, stream>>>(F2, fc_w, fc_b, (float*)d_out, NP);
}